// CTCLoss_42322607735410
// MI455X (gfx1250) — compile-verified
//
#include <hip/hip_runtime.h>
#include <hip/hip_bf16.h>
#include <stdint.h>

// Problem constants (match reference)
#define B_ 256
#define T_ 512
#define C_ 256
#define L_ 128
#define S_ 257          // 2L+1
#define BLANK_ 255      // C-1
#define EPS_ 1e-7f
#define NEG_INF_ -1.0e30f

#define NBUF_ 4         // row ring buffers
#define DIST_ 3         // async prefetch distance (NBUF_ >= DIST_+1)

// Generic pointer to a __shared__ object: low 32 bits are the workgroup-relative
// LDS byte offset (LDS aperture keeps the offset in addr[31:0]).
__device__ __forceinline__ uint32_t lds_off(const void* p) {
    return (uint32_t)(uintptr_t)p;
}

// CDNA5 async global->LDS copy, GV addressing mode (64-bit VGPR address, no SADDR).
// Each enabled lane moves 4 bytes from the global address in vaddr to the LDS address in vdst. Tracked with ASYNCcnt.
__device__ __forceinline__ void async_load_f32_to_lds(uint32_t lds_byte, const float* gp) {
    asm volatile("global_load_async_to_lds_b32 %0, %1, off"
                 :
                 : "v"(lds_byte), "v"(gp)
                 : "memory");
}

// Async loads complete in order: <=2 outstanding  =>  everything older is in LDS.
__device__ __forceinline__ void wait_async_le2() {
    asm volatile("s_wait_asynccnt 0x2" ::: "memory");
}

__device__ __forceinline__ float lse3(float a, float b, float c) {
    float m = fmaxf(a, fmaxf(b, c));
    return m + __logf(__expf(a - m) + __expf(b - m) + __expf(c - m));
}

__global__ __launch_bounds__(256)
void ctc_fwd_kernel(const int* __restrict__ y_true,
                    const float* __restrict__ y_pred,
                    const int* __restrict__ label_len,
                    float* __restrict__ per_b_loss) {
    __shared__ float row[NBUF_][C_];   // ring of softmax rows, async-filled
    __shared__ float alphaA[S_ + 2];   // [0..1] = NEG_INF pads, state s at [2+s]
    __shared__ float alphaB[S_ + 2];

    const int b   = blockIdx.x;
    const int tid = threadIdx.x;
    const int s   = tid;               // state owned by this thread; tid 0 also owns s=256

    // Per-thread extended-label metadata:
    // ext[s] = BLANK for even s, y_true[s>>1] for odd s.
    // allow_skip[s] = odd s && (s==1 || label != previous label).
    int  lbl;
    bool allow;
    if (s & 1) {
        const int k = s >> 1;
        const int c = y_true[b * L_ + k];
        lbl   = c;
        allow = (k == 0) || (c != y_true[b * L_ + k - 1]);
    } else {
        lbl   = BLANK_;
        allow = false;
    }

    if (tid == 0) {
        alphaA[0] = NEG_INF_; alphaA[1] = NEG_INF_;
        alphaB[0] = NEG_INF_; alphaB[1] = NEG_INF_;
    }

    float* cur = alphaA;
    float* nxt = alphaB;

    const float* gbase = y_pred + (size_t)b * T_ * C_;

    // Prime the pipeline: rows 0..DIST_-1 in flight (one async instr per wave each;
    // 8 waves x 32 lanes x 4B cover the 1KB row).
    for (int p = 0; p < DIST_; ++p) {
        async_load_f32_to_lds(lds_off(&row[p][tid]), gbase + (size_t)p * C_ + tid);
    }

    for (int t = 0; t < T_; ++t) {
        wait_async_le2();     // exactly DIST_ loads outstanding at loop top
                              // => this wave's row-t data has landed in LDS
        __syncthreads();      // all waves' row-t data + alpha(t-1) writes visible

        // Keep DIST_ loads in flight. Source row clamped to T-1 near the tail so
        // the outstanding count (and the wait immediate) stays uniform; the few
        // clamped duplicates land in buffers that are never read again.
        {
            const int tf = (t + DIST_ < T_) ? (t + DIST_) : (T_ - 1);
            async_load_f32_to_lds(lds_off(&row[(t + DIST_) & (NBUF_ - 1)][tid]),
                                  gbase + (size_t)tf * C_ + tid);
        }

        const float* r  = row[t & (NBUF_ - 1)];
        const float  lp = __logf(r[lbl] + EPS_);   // LDS gather + hw log

        float nv, nv2 = NEG_INF_;
        if (t == 0) {
            nv = (s < 2) ? lp : NEG_INF_;          // init: alpha0 at s=0,1 only
        } else {
            const float a0 = cur[2 + s];
            const float a1 = cur[1 + s];
            const float a2 = allow ? cur[s] : NEG_INF_;
            nv = lse3(a0, a1, a2) + lp;
            if (tid == 0) {
                // state 256 is a blank; shares lp with state 0 (also blank)
                const float b0 = cur[2 + 256];
                const float b1 = cur[1 + 256];
                nv2 = lse3(b0, b1, NEG_INF_) + lp;
            }
        }
        nxt[2 + s] = nv;
        if (tid == 0) nxt[2 + 256] = nv2;

        float* tmp = cur; cur = nxt; nxt = tmp;
    }

    __syncthreads();
    if (tid == 0) {
        const int   len    = label_len[b];            // in [1, L]
        const float a_last = cur[2 + 2 * len];
        const float a_prev = cur[2 + 2 * len - 1];
        const float m      = fmaxf(a_last, a_prev);
        const float ll     = m + __logf(__expf(a_last - m) + __expf(a_prev - m));
        per_b_loss[b] = -ll;
    }
}

// Deterministic fixed-order tree reduction of the 256 per-batch losses.
__global__ __launch_bounds__(256)
void ctc_reduce_kernel(const float* __restrict__ per_b_loss, float* __restrict__ out) {
    __shared__ float sm[256];
    const int tid = threadIdx.x;
    sm[tid] = per_b_loss[tid];
    __syncthreads();
    for (int off = 128; off > 0; off >>= 1) {
        if (tid < off) sm[tid] += sm[tid + off];
        __syncthreads();
    }
    if (tid == 0) out[0] = sm[0] * (1.0f / (float)B_);
}

extern "C" void kernel_launch(void* const* d_in, const int* in_sizes, int n_in,
                              void* d_out, int out_size, void* d_ws, size_t ws_size,
                              hipStream_t stream) {
    const int*   y_true    = (const int*)d_in[0];    // (B, L) int32
    const float* y_pred    = (const float*)d_in[1];  // (B, T, C) float32
    const int*   label_len = (const int*)d_in[2];    // (B, 1) int32
    float*       ws        = (float*)d_ws;           // B floats of scratch
    float*       out       = (float*)d_out;          // scalar float32

    ctc_fwd_kernel<<<B_, 256, 0, stream>>>(y_true, y_pred, label_len, ws);
    ctc_reduce_kernel<<<1, 256, 0, stream>>>(ws, out);
}